// MultiScaleRetention_50096498540555
// MI455X (gfx1250) — compile-verified
//
#include <hip/hip_runtime.h>
#include <hip/hip_bf16.h>
#include <math.h>

// Problem constants (match reference)
#define HIDC 1024
#define NHC  16
#define HDC  64
#define BBC  2
#define SSC  2048

typedef float v2f __attribute__((ext_vector_type(2)));
typedef float v8f __attribute__((ext_vector_type(8)));

// D = A(16x4, f32) * B(4x16, f32) + C(16x16, f32)  -- CDNA5 f32 WMMA
__device__ __forceinline__ v8f wmma4(v2f a, v2f b, v8f c) {
    return __builtin_amdgcn_wmma_f32_16x16x4_f32(
        /*neg_a=*/false, a, /*neg_b=*/false, b,
        /*c_mod=*/(short)0, c, /*reuse_a=*/false, /*reuse_b=*/false);
}

// Contiguous, 8-byte-aligned f32 pair -> single global_load_b64
__device__ __forceinline__ v2f load2(const float* __restrict__ p) {
    return *(const v2f*)p;
}

// A-matrix 16x4 fragment from row-major A (lda): lanes 0-15 M=0..15 K=k0..k0+1,
// lanes 16-31 same M, K=k0+2..k0+3.  (contiguous pair -> vector load)
__device__ __forceinline__ v2f load_a16x4(const float* __restrict__ A, int lda, int k0, int m, int kb2) {
    return load2(A + m * lda + k0 + kb2);
}

// B-matrix 4x16 fragment from row-major B[K][N] (ldb): n = lane&15; rows kb2, kb2+1.
__device__ __forceinline__ v2f load_b4x16(const float* __restrict__ B, int ldb, int k0, int n0,
                                          int nl, int kb2) {
    const float* p = B + (k0 + kb2) * ldb + n0 + nl;
    v2f r; r[0] = p[0]; r[1] = p[ldb];
    return r;
}

// ---------------------------------------------------------------------------
// Kernel 1: per-head QKV projections. One wave per (16-row strip, head).
//   x : [B*S, HID];  wq/wk/wv[n] : [HD, HD] row-major
//   q,k out: [B, NH, S, HD];  v out TRANSPOSED: [B, NH, HD, S]
// ---------------------------------------------------------------------------
__global__ void __launch_bounds__(32)
qkv_kernel(const float* __restrict__ x,
           const float* __restrict__ wq, const float* __restrict__ wk,
           const float* __restrict__ wv,
           float* __restrict__ q, float* __restrict__ k, float* __restrict__ vT)
{
    const int lane = threadIdx.x;
    const int m    = lane & 15;
    const int hi   = lane >> 4;
    const int kb2  = hi << 1;
    const int strip = blockIdx.x;          // 0 .. B*S/16 - 1
    const int n     = blockIdx.y;          // head
    const int r0    = strip * 16;          // flat row in [B*S]

    const float* Ab = x + (size_t)r0 * HIDC + n * HDC;
    const float* Wq = wq + n * HDC * HDC;
    const float* Wk = wk + n * HDC * HDC;
    const float* Wv = wv + n * HDC * HDC;

    v8f aq[4] = {}; v8f ak[4] = {}; v8f av[4] = {};
    for (int k0 = 0; k0 < HDC; k0 += 4) {
        v2f a = load_a16x4(Ab, HIDC, k0, m, kb2);
#pragma unroll
        for (int nt = 0; nt < 4; ++nt) {
            v2f bq = load_b4x16(Wq, HDC, k0, nt * 16, m, kb2);
            aq[nt] = wmma4(a, bq, aq[nt]);
            v2f bk = load_b4x16(Wk, HDC, k0, nt * 16, m, kb2);
            ak[nt] = wmma4(a, bk, ak[nt]);
            v2f bv = load_b4x16(Wv, HDC, k0, nt * 16, m, kb2);
            av[nt] = wmma4(a, bv, av[nt]);
        }
    }

    const int b  = r0 / SSC;
    const int s0 = r0 % SSC;
    const int mbase = hi * 8;              // C layout: VGPR j -> M = mbase + j
    float* qb = q  + (((size_t)b * NHC + n) * SSC + s0) * HDC;
    float* kb = k  + (((size_t)b * NHC + n) * SSC + s0) * HDC;
    float* vb = vT + (((size_t)b * NHC + n) * HDC) * SSC;   // [HD][S]
#pragma unroll
    for (int nt = 0; nt < 4; ++nt)
#pragma unroll
        for (int j = 0; j < 8; ++j) {
            size_t off = (size_t)(mbase + j) * HDC + nt * 16 + m;
            qb[off] = aq[nt][j];
            kb[off] = ak[nt][j];
            // transposed: element (row s0+mbase+j, col e=nt*16+m) -> vT[e][s]
            vb[(size_t)(nt * 16 + m) * SSC + s0 + mbase + j] = av[nt][j];
        }
}

// ---------------------------------------------------------------------------
// Kernel 2: causal decayed retention. One wave per (b, head, s-tile of 16).
//   ret[b,s,n,:] = sum_{t<=s} (q_s . k_t) * gamma_n^(s-t) * v_t
//   q,k: [B, NH, S, HD];  vT: [B, NH, HD, S];  ret out: [B, S, NH, HD]
// ---------------------------------------------------------------------------
__global__ void __launch_bounds__(32)
retention_kernel(const float* __restrict__ q, const float* __restrict__ k,
                 const float* __restrict__ vT, float* __restrict__ ret)
{
    const int lane = threadIdx.x;
    const int m    = lane & 15;
    const int hi   = lane >> 4;
    const int kb2  = hi << 1;
    const int mbase = hi * 8;
    const int st   = blockIdx.x;           // s-tile: 0 .. S/16-1
    const int bn   = blockIdx.y;           // 0 .. B*NH-1
    const int b    = bn / NHC;
    const int n    = bn % NHC;

    const float* Qb = q  + (((size_t)b * NHC + n) * SSC) * HDC;
    const float* Kb = k  + (((size_t)b * NHC + n) * SSC) * HDC;
    const float* Vb = vT + (((size_t)b * NHC + n) * HDC) * SSC;

    // gamma_n = 1 - exp(linspace(log(1/512), log(1/32), NH))[n]
    const float lin   = logf(1.0f / 512.0f)
                      + (float)n * ((logf(1.0f / 32.0f) - logf(1.0f / 512.0f)) / 15.0f);
    const float l2g   = log2f(1.0f - expf(lin));

    __shared__ float sS[16][18];           // stride 18: 8B-aligned pairs, conflict-free

    // Hoist the Q tile (A-fragments) into registers: 16 x v2f = 32 VGPRs.
    const float* Qt = Qb + st * 16 * HDC;
    v2f qa[16];
#pragma unroll
    for (int kk = 0; kk < 16; ++kk)
        qa[kk] = load2(Qt + m * HDC + kk * 4 + kb2);

    v8f acc[4] = {};
    for (int tt = 0; tt <= st; ++tt) {
        const float* Kt = Kb + tt * 16 * HDC;
        if (tt < st) {
            // Next K tile = 16*64 f32 = 4KB: one cacheline per lane.
            __builtin_prefetch((const char*)(Kt + 16 * HDC) + lane * 128, 0, 1);
        }
        // --- scores(16x16) = Q_tile(16x64) x K_tile^T(64x16) ---
        // B element (kk, t) = K[t0+t][kk]: contiguous pair per lane (t = lane&15).
        v8f sc = {};
#pragma unroll
        for (int kk = 0; kk < 16; ++kk) {
            v2f bb = load2(Kt + m * HDC + kk * 4 + kb2);
            sc = wmma4(qa[kk], bb, sc);
        }
        // --- causal decay mask on C layout, stage via LDS (C->A relayout) ---
#pragma unroll
        for (int j = 0; j < 8; ++j) {
            int sg = st * 16 + mbase + j;
            int tg = tt * 16 + m;
            float d = (float)(sg - tg);
            sS[mbase + j][m] = (sg >= tg) ? sc[j] * exp2f(d * l2g) : 0.0f;
        }
        __syncthreads();
        // --- ret(16x64) += scores(16x16) x V_tile(16x64) ---
        // B element (k, e) = vT[e0+e][t0+k]: contiguous pair per lane.
#pragma unroll
        for (int k0 = 0; k0 < 16; k0 += 4) {
            v2f a2 = *(const v2f*)&sS[m][k0 + kb2];
#pragma unroll
            for (int nt = 0; nt < 4; ++nt) {
                v2f b2 = load2(Vb + (size_t)(nt * 16 + m) * SSC + tt * 16 + k0 + kb2);
                acc[nt] = wmma4(a2, b2, acc[nt]);
            }
        }
        __syncthreads();
    }

    // store ret as [B, S, NH, HD]
    float* rb = ret + (((size_t)b * SSC + st * 16) * NHC + n) * HDC;
#pragma unroll
    for (int nt = 0; nt < 4; ++nt)
#pragma unroll
        for (int j = 0; j < 8; ++j)
            rb[(size_t)(mbase + j) * NHC * HDC + nt * 16 + m] = acc[nt][j];
}

// ---------------------------------------------------------------------------
// Kernel 3: GroupNorm (per head group of 64 channels) + per-channel affine.
//   ret: [B, S, NH, HD] == [B, S, HID];  norm out: [B, S, HID]
// ---------------------------------------------------------------------------
__global__ void groupnorm_kernel(const float* __restrict__ ret,
                                 const float* __restrict__ scale,
                                 const float* __restrict__ bias,
                                 float* __restrict__ normout)
{
    int idx = blockIdx.x * blockDim.x + threadIdx.x;   // over B*S*NH
    if (idx >= BBC * SSC * NHC) return;
    int n = idx % NHC;
    const float* p = ret + (size_t)idx * HDC;

    float mean = 0.0f;
    for (int e = 0; e < HDC; ++e) mean += p[e];
    mean *= (1.0f / HDC);
    float var = 0.0f;
    for (int e = 0; e < HDC; ++e) { float d = p[e] - mean; var += d * d; }
    var *= (1.0f / HDC);
    float inv = rsqrtf(var + 1e-5f);

    float* o = normout + (size_t)idx * HDC;
    for (int e = 0; e < HDC; ++e) {
        int c = n * HDC + e;
        o[e] = (p[e] - mean) * inv * scale[c] + bias[c];
    }
}

// ---------------------------------------------------------------------------
// Kernels 4/5: [B*S,1024] x [1024,1024] GEMM.
//   One wave per (16-row, 128-col) tile: 8 accumulator tiles, so each A
//   fragment feeds 8 WMMAs (better A reuse, fewer redundant loads).
//   SWISH_ADD: out = g*sigmoid(g) + addn  (gate branch); else plain store.
// ---------------------------------------------------------------------------
template <bool SWISH_ADD>
__global__ void __launch_bounds__(32)
gemm1024_kernel(const float* __restrict__ A, const float* __restrict__ W,
                const float* __restrict__ addn, float* __restrict__ out)
{
    const int lane = threadIdx.x;
    const int m    = lane & 15;
    const int hi   = lane >> 4;
    const int kb2  = hi << 1;
    const int strip = blockIdx.x;          // M/16
    const int nb    = blockIdx.y;          // N/128

    const float* Ab = A + (size_t)strip * 16 * HIDC;
    v8f acc[8] = {};
    for (int k0 = 0; k0 < HIDC; k0 += 4) {
        v2f a = load_a16x4(Ab, HIDC, k0, m, kb2);
#pragma unroll
        for (int nt = 0; nt < 8; ++nt) {
            v2f bb = load_b4x16(W, HIDC, k0, nb * 128 + nt * 16, m, kb2);
            acc[nt] = wmma4(a, bb, acc[nt]);
        }
    }

    const int mbase = hi * 8;
#pragma unroll
    for (int nt = 0; nt < 8; ++nt)
#pragma unroll
        for (int j = 0; j < 8; ++j) {
            size_t off = (size_t)(strip * 16 + mbase + j) * HIDC + nb * 128 + nt * 16 + m;
            float g = acc[nt][j];
            if constexpr (SWISH_ADD) {
                out[off] = g / (1.0f + expf(-g)) + addn[off];   // swish(g) + norm
            } else {
                out[off] = g;
            }
        }
}

// ---------------------------------------------------------------------------
extern "C" void kernel_launch(void* const* d_in, const int* in_sizes, int n_in,
                              void* d_out, int out_size, void* d_ws, size_t ws_size,
                              hipStream_t stream)
{
    (void)in_sizes; (void)n_in; (void)out_size; (void)ws_size;
    const float* x   = (const float*)d_in[0];
    const float* wq  = (const float*)d_in[1];
    const float* wk  = (const float*)d_in[2];
    const float* wv  = (const float*)d_in[3];
    const float* w1  = (const float*)d_in[4];
    const float* w2  = (const float*)d_in[5];
    const float* gns = (const float*)d_in[6];
    const float* gnb = (const float*)d_in[7];
    float* out = (float*)d_out;
    float* ws  = (float*)d_ws;

    const size_t NQ = (size_t)BBC * NHC * SSC * HDC;   // 4M floats = 16 MB
    float* q    = ws;
    float* k    = ws + NQ;
    float* vT   = ws + 2 * NQ;
    float* ret  = ws + 3 * NQ;
    float* norm = q;   // q dead after retention
    float* mid  = k;   // k dead after retention
    // total workspace use: 4 * 16 MB = 64 MB

    dim3 wave(32);
    qkv_kernel<<<dim3(BBC * SSC / 16, NHC), wave, 0, stream>>>(x, wq, wk, wv, q, k, vT);
    retention_kernel<<<dim3(SSC / 16, BBC * NHC), wave, 0, stream>>>(q, k, vT, ret);
    groupnorm_kernel<<<dim3(BBC * SSC * NHC / 256), dim3(256), 0, stream>>>(ret, gns, gnb, norm);
    gemm1024_kernel<true ><<<dim3(BBC * SSC / 16, HIDC / 128), wave, 0, stream>>>(x,   w1, norm, mid);
    gemm1024_kernel<false><<<dim3(BBC * SSC / 16, HIDC / 128), wave, 0, stream>>>(mid, w2, nullptr, out);
}